// LearnPartitioning_45226005627027
// MI455X (gfx1250) — compile-verified
//
#include <hip/hip_runtime.h>
#include <stdint.h>

#define GLOBAL_AS __attribute__((address_space(1)))
#define LDS_AS    __attribute__((address_space(3)))

typedef float v4f __attribute__((ext_vector_type(4)));
typedef int   v4i __attribute__((ext_vector_type(4)));

namespace {
constexpr int   Cc = 8, Hh = 320, Ww = 320, Bb = 4, NCOIL = 12;
constexpr int   HW = Hh * Ww;                 // 102400
constexpr int   PLANES = Bb * Cc;             // 32
constexpr int   TPB = 256;
constexpr int   VECS = HW / 4;                // 25600 float4 per (b,c) plane
constexpr int   BLKS_PER_PLANE = VECS / TPB;  // 100
constexpr float SLOPE_P   = 5.0f;
constexpr float PTOT      = (float)HW / 4.0f - 100.0f;  // H*W/R - K^2 = 25500
constexpr float INV_TOTAL = (float)HW * 0.75f;          // H*W*(1-1/R) = 76800
constexpr int   C_LO = 155, C_HI = 165;                 // 10x10 center region
}

// ---- gfx1250 async global->LDS path (ASYNCcnt-tracked) ----
// Builtin signature (confirmed by clang): (v4i AS1*, v4i AS3*, imm offset, imm cpol).
#if defined(__has_builtin)
#  if __has_builtin(__builtin_amdgcn_global_load_async_to_lds_b128)
#    define ASYNC_COPY_B128(gp, lp)                                   \
       __builtin_amdgcn_global_load_async_to_lds_b128(                \
           (GLOBAL_AS v4i*)(uintptr_t)(const void*)(gp),              \
           (LDS_AS v4i*)(lp), 0, 0)
#    define HAVE_ASYNC_LDS 1
#  endif
#endif

#if defined(HAVE_ASYNC_LDS)
#  if defined(__has_builtin) && __has_builtin(__builtin_amdgcn_s_wait_asynccnt)
#    define WAIT_ASYNC0()                         \
       do {                                       \
         __builtin_amdgcn_s_wait_asynccnt(0);     \
         asm volatile("" ::: "memory");           \
       } while (0)
#  else
#    define WAIT_ASYNC0() asm volatile("s_wait_asynccnt 0" ::: "memory")
#  endif
#endif

// Fast reciprocal: v_rcp_f32 instead of the IEEE div_scale/div_fmas chain.
__device__ __forceinline__ float fast_rcp(float x) {
#if defined(__has_builtin) && __has_builtin(__builtin_amdgcn_rcpf)
  return __builtin_amdgcn_rcpf(x);
#else
  return 1.0f / x;
#endif
}

__device__ __forceinline__ float sigm5(float x) {
  // sigmoid(5x) = 1 / (1 + exp(-5x)); v_exp_f32 + v_rcp_f32
  return fast_rcp(1.0f + __expf(-SLOPE_P * x));
}

// ---------------------------------------------------------------------------
// Kernel A: per-channel masked probability sum. 8 blocks (one per channel).
// ---------------------------------------------------------------------------
__global__ __launch_bounds__(TPB) void psum_kernel(const float* __restrict__ wts,
                                                   float* __restrict__ psum) {
  __shared__ float red[TPB];
  const int c = blockIdx.x;
  float acc = 0.0f;
  for (int i = threadIdx.x; i < HW; i += TPB) {
    const int h = i / Ww;
    const int w = i % Ww;
    const bool cen = (h >= C_LO) & (h < C_HI) & (w >= C_LO) & (w < C_HI);
    const float p = sigm5(wts[(size_t)c * HW + i]);
    acc += cen ? 0.0f : p;
  }
  red[threadIdx.x] = acc;
  __syncthreads();
  #pragma unroll
  for (int s = TPB / 2; s > 0; s >>= 1) {
    if (threadIdx.x < s) red[threadIdx.x] += red[threadIdx.x + s];
    __syncthreads();
  }
  if (threadIdx.x == 0) psum[c] = red[0];
}

// ---------------------------------------------------------------------------
// Kernel B: fused norm_prob + threshold + coil-broadcast partition.
// One block = 256 float4 positions of one (b,c) plane; loops 12 coils.
// initial_mask streamed via async global->LDS b128 DMA (overlapped with the
// sigmoid/threshold VALU work), outputs written with non-temporal b128 stores.
// ---------------------------------------------------------------------------
__global__ __launch_bounds__(TPB) void fused_partition_kernel(
    const float* __restrict__ mask, const float* __restrict__ wts,
    const float* __restrict__ noise, const float* __restrict__ psum,
    float* __restrict__ out) {
#if defined(HAVE_ASYNC_LDS)
  __shared__ float smem[NCOIL * TPB * 4];  // 48 KB: 12 coil tiles of 256 float4
#endif
  const int plane = blockIdx.x / BLKS_PER_PLANE;                 // b*C + c
  const int vidx  = (blockIdx.x % BLKS_PER_PLANE) * TPB + threadIdx.x;
  const int c     = plane % Cc;
  const int e     = vidx * 4;       // flat (h,w) element base, 4 per thread
  const int h     = e / Ww;
  const int wc    = e % Ww;

  const size_t mbase = (size_t)plane * NCOIL * HW + e;

#if defined(HAVE_ASYNC_LDS)
  // Issue all 12 coil-tile DMAs up front; overlap with lam computation below.
  {
    float* lp = &smem[threadIdx.x * 4];
    #pragma unroll
    for (int k = 0; k < NCOIL; ++k)
      ASYNC_COPY_B128(mask + mbase + (size_t)k * HW, lp + k * (TPB * 4));
  }
#endif

  // Channel-uniform rescale: np = p*scl + off  (branch on psum vs ptot).
  const float ps   = psum[c];
  const float fup  = INV_TOTAL / ((float)HW - ps - 100.0f);
  const bool  down = ps > PTOT;
  const float scl  = down ? (PTOT / ps) : fup;
  const float off  = down ? 0.0f : (1.0f - fup);

  const v4f wv = *reinterpret_cast<const v4f*>(wts + (size_t)c * HW + e);
  const v4f nv = *reinterpret_cast<const v4f*>(noise + (size_t)plane * HW + e);

  const bool rowc = (h >= C_LO) & (h < C_HI);
  bool lam[4];
  #pragma unroll
  for (int j = 0; j < 4; ++j) {
    const bool cen = rowc & (wc + j >= C_LO) & (wc + j < C_HI);
    const float p  = sigm5(wv[j]);
    const float np = cen ? 1.0f : fmaf(p, scl, off);
    lam[j] = (np - nv[j]) > 0.0f;
  }

  const size_t TOT = (size_t)PLANES * NCOIL * HW;  // elements per output tensor

#if defined(HAVE_ASYNC_LDS)
  WAIT_ASYNC0();  // this wave's DMAs landed in LDS; lanes read only their own slots
#endif

  #pragma unroll
  for (int k = 0; k < NCOIL; ++k) {
#if defined(HAVE_ASYNC_LDS)
    const v4f m = *reinterpret_cast<const v4f*>(&smem[threadIdx.x * 4 + k * (TPB * 4)]);
#else
    const v4f m = *reinterpret_cast<const v4f*>(mask + mbase + (size_t)k * HW);
#endif
    v4f o1, o2;
    #pragma unroll
    for (int j = 0; j < 4; ++j) {
      o1[j] = lam[j] ? m[j] : 0.0f;
      o2[j] = lam[j] ? 0.0f : m[j];
    }
    const size_t o = mbase + (size_t)k * HW;
    __builtin_nontemporal_store(o1, reinterpret_cast<v4f*>(out + o));
    __builtin_nontemporal_store(o2, reinterpret_cast<v4f*>(out + TOT + o));
  }
}

// ---------------------------------------------------------------------------
extern "C" void kernel_launch(void* const* d_in, const int* in_sizes, int n_in,
                              void* d_out, int out_size, void* d_ws, size_t ws_size,
                              hipStream_t stream) {
  (void)in_sizes; (void)n_in; (void)out_size; (void)ws_size;
  const float* mask  = (const float*)d_in[0];  // (B,C,COILS,H,W)
  const float* wts   = (const float*)d_in[1];  // (C,H,W)
  const float* noise = (const float*)d_in[2];  // (B,C,H,W)
  float* out  = (float*)d_out;                 // 2 x (B,C,COILS,H,W), concat
  float* psum = (float*)d_ws;                  // 8 floats

  psum_kernel<<<Cc, TPB, 0, stream>>>(wts, psum);
  fused_partition_kernel<<<PLANES * BLKS_PER_PLANE, TPB, 0, stream>>>(
      mask, wts, noise, psum, out);
}